// SEModule_10788957847763
// MI455X (gfx1250) — compile-verified
//
#include <hip/hip_runtime.h>
#include <math.h>

#define B_   64
#define C_   512
#define HW_  3136
#define HW4_ 784
#define CR_  32
#define EPS_ 1e-5f

typedef float v2f __attribute__((ext_vector_type(2)));
typedef float v4f __attribute__((ext_vector_type(4)));
typedef float v8f __attribute__((ext_vector_type(8)));

// ---------------------------------------------------------------------------
// Kernel 1: squeeze — global average pool over H*W per (b,c).
// One block per (b,c) plane (12.5 KB), float4 streaming + LDS tree reduce.
// ---------------------------------------------------------------------------
__global__ void se_squeeze(const float* __restrict__ x, float* __restrict__ s) {
    __shared__ float red[256];
    const int bc = blockIdx.x;
    const v4f* xp = (const v4f*)x + (size_t)bc * HW4_;
    float sum = 0.f;
    for (int i = threadIdx.x; i < HW4_; i += 256) {
        v4f v = xp[i];
        sum += (v.x + v.y) + (v.z + v.w);
    }
    red[threadIdx.x] = sum;
    __syncthreads();
    for (int off = 128; off > 0; off >>= 1) {
        if ((int)threadIdx.x < off) red[threadIdx.x] += red[threadIdx.x + off];
        __syncthreads();
    }
    if (threadIdx.x == 0) s[bc] = red[0] * (1.0f / (float)HW_);
}

// ---------------------------------------------------------------------------
// Kernel 2: gate — fc1+BN+ReLU, fc2+BN+sigmoid, all via V_WMMA_F32_16X16X4_F32.
// Single 256-thread block (8 waves). h staged through LDS between GEMMs.
// A-matrix layout (16x4 f32): lane<16 holds {K=0,K=1}, lane>=16 holds {K=2,K=3}
// of row M = lane%16. C/D layout: elem i of v8f = (M = i + 8*(lane>=16),
// N = lane%16) -> BN scale/bias is per-lane uniform across the accumulator.
// ---------------------------------------------------------------------------
__global__ void se_gate(const float* __restrict__ s,
                        const float* __restrict__ w1,
                        const float* __restrict__ g1, const float* __restrict__ b1,
                        const float* __restrict__ m1, const float* __restrict__ v1,
                        const float* __restrict__ w2,
                        const float* __restrict__ g2, const float* __restrict__ b2,
                        const float* __restrict__ m2, const float* __restrict__ v2,
                        float* __restrict__ gate) {
    __shared__ float hbuf[B_ * CR_];   // 64 x 32 fp32 = 8 KB
    const int tid  = threadIdx.x;
    const int wave = tid >> 5;
    const int lane = tid & 31;
    const int lo   = lane & 15;            // N (or M for A) within tile
    const int hi8  = (lane >> 4) << 3;     // +8 row offset for upper half-wave
    const int koff = (lane >> 4) * 2;      // K sub-offset for A/B operands

    // ---- GEMM1: h[64x32] = s[64x512] @ w1[32x512]^T, one 16x16 tile per wave
    {
        const int Mb = (wave >> 1) * 16;
        const int Nb = (wave & 1) * 16;
        const float* arow = s  + (size_t)(Mb + lo) * C_;
        const float* brow = w1 + (size_t)(Nb + lo) * C_;
        v8f acc = {};
        for (int k = 0; k < C_; k += 4) {
            v2f a, b;
            a.x = arow[k + koff]; a.y = arow[k + koff + 1];
            b.x = brow[k + koff]; b.y = brow[k + koff + 1];
            acc = __builtin_amdgcn_wmma_f32_16x16x4_f32(
                false, a, false, b, (short)0, acc, false, false);
        }
        const int r   = Nb + lo;                       // CR channel
        const float sc = g1[r] * rsqrtf(v1[r] + EPS_);
        const float bi = b1[r] - m1[r] * sc;
#pragma unroll
        for (int i = 0; i < 8; ++i) {
            float h = fmaxf(acc[i] * sc + bi, 0.0f);   // BN1 + ReLU
            hbuf[(Mb + i + hi8) * CR_ + r] = h;
        }
    }
    __syncthreads();

    // ---- GEMM2: g[64x512] = h[64x32] @ w2[512x32]^T, 128 tiles, 16 per wave
    for (int t = 0; t < 16; ++t) {
        const int tile = t * 8 + wave;
        const int Mb = (tile & 3) * 16;
        const int Nb = (tile >> 2) * 16;
        const float* arow = hbuf + (Mb + lo) * CR_;
        const float* brow = w2 + (size_t)(Nb + lo) * CR_;
        v8f acc = {};
#pragma unroll
        for (int k = 0; k < CR_; k += 4) {
            v2f a, b;
            a.x = arow[k + koff]; a.y = arow[k + koff + 1];
            b.x = brow[k + koff]; b.y = brow[k + koff + 1];
            acc = __builtin_amdgcn_wmma_f32_16x16x4_f32(
                false, a, false, b, (short)0, acc, false, false);
        }
        const int c   = Nb + lo;                       // output channel
        const float sc = g2[c] * rsqrtf(v2[c] + EPS_);
        const float bi = b2[c] - m2[c] * sc;
#pragma unroll
        for (int i = 0; i < 8; ++i) {
            float gv = acc[i] * sc + bi;               // BN2
            gv = 1.0f / (1.0f + __expf(-gv));          // sigmoid
            gate[(size_t)(Mb + i + hi8) * C_ + c] = gv;
        }
    }
}

// ---------------------------------------------------------------------------
// Kernel 3: excite — out = x * g[b,c]; float4 streams, NT stores (411 MB,
// never re-read, keep it out of L2).
// ---------------------------------------------------------------------------
__global__ void se_excite(const float* __restrict__ x,
                          const float* __restrict__ gate,
                          float* __restrict__ out) {
    const size_t i4 = (size_t)blockIdx.x * 256 + threadIdx.x;
    const int bc = (int)(i4 / HW4_);        // HW=3136 divisible by 4
    const float g = gate[bc];
    v4f v = ((const v4f*)x)[i4];
    v.x *= g; v.y *= g; v.z *= g; v.w *= g;
    __builtin_nontemporal_store(v, (v4f*)out + i4);
}

// ---------------------------------------------------------------------------
extern "C" void kernel_launch(void* const* d_in, const int* in_sizes, int n_in,
                              void* d_out, int out_size, void* d_ws, size_t ws_size,
                              hipStream_t stream) {
    const float* x  = (const float*)d_in[0];
    const float* w1 = (const float*)d_in[1];
    const float* g1 = (const float*)d_in[2];
    const float* b1 = (const float*)d_in[3];
    const float* m1 = (const float*)d_in[4];
    const float* v1 = (const float*)d_in[5];
    const float* w2 = (const float*)d_in[6];
    const float* g2 = (const float*)d_in[7];
    const float* b2 = (const float*)d_in[8];
    const float* m2 = (const float*)d_in[9];
    const float* v2 = (const float*)d_in[10];
    float* out  = (float*)d_out;

    float* s    = (float*)d_ws;          // [64*512] pooled means
    float* gate = s + B_ * C_;           // [64*512] sigmoid gates

    se_squeeze<<<B_ * C_, 256, 0, stream>>>(x, s);
    se_gate<<<1, 256, 0, stream>>>(s, w1, g1, b1, m1, v1,
                                   w2, g2, b2, m2, v2, gate);
    const int n4      = (B_ * C_ * HW_) / 4;     // 25,690,112 float4
    const int nblocks = n4 / 256;                // 100,352 (exact)
    se_excite<<<nblocks, 256, 0, stream>>>(x, gate, out);
}